// RetinaFaceFishBBoxLoss_30262339567894
// MI455X (gfx1250) — compile-verified
//
#include <hip/hip_runtime.h>
#include <hip/hip_bf16.h>
#include <stdint.h>

// Problem constants (match the reference).
#define B_DIM   32
#define P_DIM   67200
#define NEGPOS  7

typedef float v2f __attribute__((ext_vector_type(2)));
typedef float v4f __attribute__((ext_vector_type(4)));
typedef float v8f __attribute__((ext_vector_type(8)));

__device__ __forceinline__ float smooth_l1(float x) {
    float a = fabsf(x);
    return (a < 1.0f) ? 0.5f * x * x : (a - 0.5f);
}

// Wave32-wide sum using the matrix pipe:
// A is the 16x4 f32 operand (2 VGPRs/lane): VGPR0 = v, VGPR1 = 0.
// B = all-ones 4x16 (layout-invariant). D[i][j] = v_i + v_{i+16}.
// Each lane sums its 8 D VGPRs (rows 0..7 or 8..15), then one cross-half
// shuffle-add gives the full 32-lane total in every lane.
__device__ __forceinline__ float wave_sum_wmma(float v) {
    v2f a; a[0] = v;    a[1] = 0.0f;
    v2f b; b[0] = 1.0f; b[1] = 1.0f;
    v8f c = {0.f, 0.f, 0.f, 0.f, 0.f, 0.f, 0.f, 0.f};
    v8f d = __builtin_amdgcn_wmma_f32_16x16x4_f32(
        /*neg_a=*/false, a, /*neg_b=*/false, b,
        /*c_mod=*/(short)0, c, /*reuse_a=*/false, /*reuse_b=*/false);
    float s = ((d[0] + d[1]) + (d[2] + d[3])) + ((d[4] + d[5]) + (d[6] + d[7]));
    s += __shfl_xor(s, 16, 32);
    return s;
}

// ---------------------------------------------------------------------------
// Kernel 0: zero the accumulator header of the workspace.
// ---------------------------------------------------------------------------
__global__ void init_ws_kernel(float* accF, int* accI, int* rowPos) {
    int t = threadIdx.x;
    if (t < 4)  accF[t] = 0.0f;
    if (t < 2)  accI[t] = 0;
    if (t < 32) rowPos[t] = 0;
}

// ---------------------------------------------------------------------------
// Kernel A: one streaming pass over all B*P anchors.
//   - masked smooth-L1 sums (loc over pos, landm over pos1) — loads skipped
//     behind the mask (only ~2% of anchors are positive)
//   - per-anchor CE, accumulated over positives
//   - writes mine = pos ? 0 : ce  (kept hot in L2 for the select kernel)
//   - per-row positive counts
// Block = 256 threads (8 waves). Wave reductions use v_wmma_f32_16x16x4_f32.
// ---------------------------------------------------------------------------
__global__ void __launch_bounds__(256) loss_stream_kernel(
    const float* __restrict__ loc_data, const float* __restrict__ conf_data,
    const float* __restrict__ landm_data, const float* __restrict__ loc_t,
    const float* __restrict__ landm_t, const int* __restrict__ conf_t,
    float* __restrict__ accF, int* __restrict__ accI,
    int* __restrict__ rowPos, float* __restrict__ mine)
{
    const int b = blockIdx.y;
    const int p = blockIdx.x * 256 + threadIdx.x;

    float sL = 0.0f, sLM = 0.0f, sCE = 0.0f;
    int isPos = 0, isPos1 = 0;

    if (p < P_DIM) {
        const size_t idx = (size_t)b * P_DIM + p;
        const int t = __builtin_nontemporal_load(conf_t + idx);
        const bool pos  = (t != 0);
        const bool pos1 = (t > 0);
        isPos = pos; isPos1 = pos1;

        if (pos) {
            v4f ld = __builtin_nontemporal_load((const v4f*)(loc_data + idx * 4));
            v4f lt = __builtin_nontemporal_load((const v4f*)(loc_t    + idx * 4));
            sL = smooth_l1(ld[0] - lt[0]) + smooth_l1(ld[1] - lt[1]) +
                 smooth_l1(ld[2] - lt[2]) + smooth_l1(ld[3] - lt[3]);
        }
        if (pos1) {
            v4f a0 = __builtin_nontemporal_load((const v4f*)(landm_data + idx * 8));
            v4f a1 = __builtin_nontemporal_load((const v4f*)(landm_data + idx * 8 + 4));
            v4f t0 = __builtin_nontemporal_load((const v4f*)(landm_t    + idx * 8));
            v4f t1 = __builtin_nontemporal_load((const v4f*)(landm_t    + idx * 8 + 4));
            sLM = smooth_l1(a0[0] - t0[0]) + smooth_l1(a0[1] - t0[1]) +
                  smooth_l1(a0[2] - t0[2]) + smooth_l1(a0[3] - t0[3]) +
                  smooth_l1(a1[0] - t1[0]) + smooth_l1(a1[1] - t1[1]) +
                  smooth_l1(a1[2] - t1[2]) + smooth_l1(a1[3] - t1[3]);
        }

        v2f cf = __builtin_nontemporal_load((const v2f*)(conf_data + idx * 2));
        const float c0 = cf[0], c1 = cf[1];
        const float m   = fmaxf(c0, c1);
        const float lse = m + log1pf(__expf(-fabsf(c0 - c1)));
        const float ce  = lse - (pos ? c1 : c0);      // ce >= 0 always
        sCE = pos ? ce : 0.0f;
        mine[idx] = pos ? 0.0f : ce;                  // regular store -> L2 resident
    }

    // ---- block reduction (all 256 threads converged here; EXEC all-ones) ----
    __shared__ float wsum[8][3];
    __shared__ int   wcnt[2];
    if (threadIdx.x == 0) { wcnt[0] = 0; wcnt[1] = 0; }
    __syncthreads();

    const float rL  = wave_sum_wmma(sL);
    const float rLM = wave_sum_wmma(sLM);
    const float rCE = wave_sum_wmma(sCE);

    const int wid  = threadIdx.x >> 5;
    const int lane = threadIdx.x & 31;
    if (lane == 0) { wsum[wid][0] = rL; wsum[wid][1] = rLM; wsum[wid][2] = rCE; }
    if (isPos)  atomicAdd(&wcnt[0], 1);
    if (isPos1) atomicAdd(&wcnt[1], 1);
    __syncthreads();

    if (threadIdx.x == 0) {
        float tL = 0.f, tLM = 0.f, tCE = 0.f;
        for (int w = 0; w < 8; ++w) { tL += wsum[w][0]; tLM += wsum[w][1]; tCE += wsum[w][2]; }
        if (tL  != 0.f) atomicAdd(&accF[0], tL);
        if (tLM != 0.f) atomicAdd(&accF[1], tLM);
        if (tCE != 0.f) atomicAdd(&accF[2], tCE);
        if (wcnt[0]) { atomicAdd(&rowPos[b], wcnt[0]); atomicAdd(&accI[0], wcnt[0]); }
        if (wcnt[1]) atomicAdd(&accI[1], wcnt[1]);
    }
}

// ---------------------------------------------------------------------------
// Kernel B: per-row top-K sum via 3-pass radix select on float bit patterns
// (all values >= 0 so uint order == float order). Row data is staged into LDS
// in tiles with gfx1250 async global->LDS loads (ASYNCcnt).
// loss_c_neg(row) = sum{v > T} + (K - cnt_gt) * T    (tie-invariant, exact)
// ---------------------------------------------------------------------------
#define SEL_TILE 4096
__global__ void __launch_bounds__(512) topk_select_kernel(
    const float* __restrict__ mine, const int* __restrict__ rowPos,
    float* __restrict__ accF)
{
    const int b   = blockIdx.x;
    const int tid = threadIdx.x;
    const int NT  = 512;

    const int  npos = rowPos[b];
    const long kl   = (long)NEGPOS * (long)npos;
    const int  K    = (int)(kl < (long)(P_DIM - 1) ? kl : (long)(P_DIM - 1));
    if (K <= 0) return;  // uniform per block

    const float* row = mine + (size_t)b * P_DIM;

    __shared__ unsigned hist[2048];
    __shared__ float    tile[SEL_TILE];
    __shared__ unsigned sBin, sKrem;
    __shared__ unsigned cntSh;
    __shared__ float    sumSh;

    unsigned prefix = 0u, prefmask = 0u;
    unsigned krem = (unsigned)K;
    const int      shifts[3] = {21, 10, 0};
    const unsigned bmask[3]  = {0x7FFu, 0x7FFu, 0x3FFu};

    for (int pass = 0; pass < 3; ++pass) {
        for (int i = tid; i < 2048; i += NT) hist[i] = 0u;
        __syncthreads();

        for (int base = 0; base < P_DIM; base += SEL_TILE) {
            const int n = min(SEL_TILE, P_DIM - base);
            // async global -> LDS staging (ASYNCcnt tracked)
            for (int i = tid; i < n; i += NT) {
                unsigned ldsAddr = (unsigned)(uintptr_t)(&tile[i]);
                const float* g = row + base + i;
                asm volatile("global_load_async_to_lds_b32 %0, %1, off"
                             :: "v"(ldsAddr), "v"(g) : "memory");
            }
            asm volatile("s_wait_asynccnt 0x0" ::: "memory");
            __syncthreads();

            const int shift = shifts[pass];
            const unsigned msk = bmask[pass];
            for (int i = tid; i < n; i += NT) {
                const unsigned u = __float_as_uint(tile[i]);
                if ((u & prefmask) == prefix)
                    atomicAdd(&hist[(u >> shift) & msk], 1u);
            }
            __syncthreads();
        }

        if (tid == 0) {
            unsigned cum = 0u, bsel = 0u;
            for (int bin = (int)bmask[pass]; bin >= 0; --bin) {
                const unsigned c = hist[bin];
                if (cum + c >= krem) { bsel = (unsigned)bin; break; }
                cum += c;
            }
            sBin = bsel; sKrem = krem - cum;
        }
        __syncthreads();
        prefix   |= sBin << shifts[pass];
        prefmask |= bmask[pass] << shifts[pass];
        krem      = sKrem;
        __syncthreads();
    }
    // prefix now holds the exact bit pattern of T = K-th largest value.

    if (tid == 0) { cntSh = 0u; sumSh = 0.0f; }
    __syncthreads();

    float    ls = 0.0f;
    unsigned lc = 0u;
    for (int base = 0; base < P_DIM; base += SEL_TILE) {
        const int n = min(SEL_TILE, P_DIM - base);
        for (int i = tid; i < n; i += NT) {
            unsigned ldsAddr = (unsigned)(uintptr_t)(&tile[i]);
            const float* g = row + base + i;
            asm volatile("global_load_async_to_lds_b32 %0, %1, off"
                         :: "v"(ldsAddr), "v"(g) : "memory");
        }
        asm volatile("s_wait_asynccnt 0x0" ::: "memory");
        __syncthreads();
        for (int i = tid; i < n; i += NT) {
            const float v = tile[i];
            if (__float_as_uint(v) > prefix) { lc++; ls += v; }
        }
        __syncthreads();
    }
    atomicAdd(&cntSh, lc);
    atomicAdd(&sumSh, ls);
    __syncthreads();
    if (tid == 0) {
        const float T = __uint_as_float(prefix);
        const float res = sumSh + (float)(K - (int)cntSh) * T;
        atomicAdd(&accF[3], res);
    }
}

// ---------------------------------------------------------------------------
// Kernel C: finalize the three scalars.
// ---------------------------------------------------------------------------
__global__ void finalize_kernel(const float* accF, const int* accI, float* out) {
    if (threadIdx.x == 0) {
        const float N  = fmaxf((float)accI[0], 1.0f);
        const float N1 = fmaxf((float)accI[1], 1.0f);
        out[0] = accF[0] / N;                 // loss_l / N
        out[1] = (accF[2] + accF[3]) / N;     // loss_c / N  (pos CE + mined neg CE)
        out[2] = accF[1] / N1;                // loss_landm / N1
    }
}

extern "C" void kernel_launch(void* const* d_in, const int* in_sizes, int n_in,
                              void* d_out, int out_size, void* d_ws, size_t ws_size,
                              hipStream_t stream) {
    const float* loc_data   = (const float*)d_in[0];
    const float* conf_data  = (const float*)d_in[1];
    const float* landm_data = (const float*)d_in[2];
    const float* loc_t      = (const float*)d_in[3];
    const float* landm_t    = (const float*)d_in[4];
    const int*   conf_t     = (const int*)d_in[5];

    // Workspace layout: [0..3] float accums (lossL, lossLM, cePos, ceNeg),
    // [4..5] int totals (numPos, numPos1), [8..39] int per-row pos counts,
    // then 256B-aligned mine[B*P] floats (8.6 MB, stays in the 192 MB L2).
    float* accF   = (float*)d_ws;
    int*   accI   = (int*)d_ws + 4;
    int*   rowPos = (int*)d_ws + 8;
    float* mine   = (float*)((char*)d_ws + 256);
    float* out    = (float*)d_out;

    hipLaunchKernelGGL(init_ws_kernel, dim3(1), dim3(64), 0, stream,
                       accF, accI, rowPos);

    dim3 gA((P_DIM + 255) / 256, B_DIM);
    hipLaunchKernelGGL(loss_stream_kernel, gA, dim3(256), 0, stream,
                       loc_data, conf_data, landm_data, loc_t, landm_t, conf_t,
                       accF, accI, rowPos, mine);

    hipLaunchKernelGGL(topk_select_kernel, dim3(B_DIM), dim3(512), 0, stream,
                       mine, rowPos, accF);

    hipLaunchKernelGGL(finalize_kernel, dim3(1), dim3(32), 0, stream,
                       accF, accI, out);
}